// AttnDecoderRNN_24197845745825
// MI455X (gfx1250) — compile-verified
//
#include <hip/hip_runtime.h>
#include <math.h>

typedef __attribute__((ext_vector_type(2))) float v2f;
typedef __attribute__((ext_vector_type(8))) float v8f;

#define V_SZ 50000
#define E_SZ 300
#define H_SZ 2048
#define L_SZ 512
#define EH_SZ (E_SZ + H_SZ)   /* 2348, divisible by 4 */

/* workspace layout (float offsets) */
#define WS_CAT1   0        /* 2348 */
#define WS_SCORES 2432     /* 512  */
#define WS_CAT2   3072     /* 2348 */
#define WS_X      5504     /* 2048 */
#define WS_GI     7680     /* 6144 */
#define WS_GH     13952    /* 6144 */
#define WS_HNEW   20224    /* 2048 */
#define WS_LOGITS 22400    /* 50000 */

// ---------------------------------------------------------------------------
// Generic bandwidth-bound matvec y[r] = dot(W[r,:], x) + bias[r] via
// V_WMMA_F32_16X16X4_F32. One wave32 handles a 16-row tile; K stepped by 4.
// A = 16x4 tile of W (f32 A layout: lanes 0-15 K=0..1, lanes 16-31 K=2..3),
// B = x[kb..kb+3] broadcast over 16 columns (VGPR0 = K0/K2, VGPR1 = K1/K3).
// act: 0 = none, 1 = relu.
// ---------------------------------------------------------------------------
__global__ void wmma_matvec_kernel(const float* __restrict__ W,
                                   const float* __restrict__ x,
                                   const float* __restrict__ bias,
                                   float* __restrict__ y,
                                   int R, int K, int act) {
  const int lane = threadIdx.x & 31;
  const int wave = (int)((blockIdx.x * blockDim.x + threadIdx.x) >> 5);
  const int rb   = wave << 4;
  if (rb >= R) return;                      // wave-uniform: EXEC stays all-1s
  const int half = lane >> 4;               // 0: lanes 0-15, 1: lanes 16-31
  const int m    = lane & 15;
  const float* __restrict__ wrow = W + (size_t)(rb + m) * (size_t)K;

  v8f c = {0.f, 0.f, 0.f, 0.f, 0.f, 0.f, 0.f, 0.f};
  for (int kb = 0; kb < K; kb += 4) {
    const int k0 = kb + (half << 1);        // 8-byte aligned (K even, kb even)
    v2f a; a.x = wrow[k0]; a.y = wrow[k0 + 1];
    v2f b; b.x = x[k0];    b.y = x[k0 + 1];
    c = __builtin_amdgcn_wmma_f32_16x16x4_f32(
            /*neg_a=*/false, a, /*neg_b=*/false, b,
            /*c_mod=*/(short)0, c, /*reuse_a=*/false, /*reuse_b=*/false);
  }

  // C/D f32 layout: VGPR g holds M=g (lanes 0-15) / M=g+8 (lanes 16-31).
  // Every column n carries the same value; lanes with n==0 write out.
  if (m == 0) {
    const int base = rb + (half << 3);
    #pragma unroll
    for (int g = 0; g < 8; ++g) {
      float v = c[g] + bias[base + g];
      if (act) v = fmaxf(v, 0.f);
      y[base + g] = v;
    }
  }
}

// cat1 = [emb[input], h0]; also seed cat2[0:E] = emb[input]
__global__ void build_cat1_kernel(const int* __restrict__ input,
                                  const float* __restrict__ emb,
                                  const float* __restrict__ hidden,
                                  float* __restrict__ cat1,
                                  float* __restrict__ cat2) {
  const int idx = input[0];
  const float* __restrict__ erow = emb + (size_t)idx * E_SZ;
  for (int i = blockIdx.x * blockDim.x + threadIdx.x; i < EH_SZ;
       i += gridDim.x * blockDim.x) {
    float v = (i < E_SZ) ? erow[i] : hidden[i - E_SZ];
    cat1[i] = v;
    if (i < E_SZ) cat2[i] = v;
  }
}

// stable softmax over 512 scores (in-place) + export attn_weights
__global__ void softmax512_kernel(float* __restrict__ s,
                                  float* __restrict__ out_attn) {
  __shared__ float sm[L_SZ];
  const int t = threadIdx.x;
  const float v = s[t];
  sm[t] = v; __syncthreads();
  for (int o = L_SZ / 2; o > 0; o >>= 1) {
    if (t < o) sm[t] = fmaxf(sm[t], sm[t + o]);
    __syncthreads();
  }
  const float mx = sm[0]; __syncthreads();
  const float e = __expf(v - mx);
  sm[t] = e; __syncthreads();
  for (int o = L_SZ / 2; o > 0; o >>= 1) {
    if (t < o) sm[t] += sm[t + o];
    __syncthreads();
  }
  const float w = e / sm[0];
  s[t] = w;
  out_attn[t] = w;
}

// cat2[E+h] = sum_l w[l] * enc[l*H + h]   (coalesced over h)
__global__ void attn_apply_kernel(const float* __restrict__ w,
                                  const float* __restrict__ enc,
                                  float* __restrict__ cat2) {
  const int h = blockIdx.x * blockDim.x + threadIdx.x;
  if (h >= H_SZ) return;
  float acc = 0.f;
  for (int l = 0; l < L_SZ; ++l)
    acc = fmaf(w[l], enc[(size_t)l * H_SZ + h], acc);
  cat2[E_SZ + h] = acc;
}

// GRU gate math (PyTorch r,z,n layout); writes h_new to ws and d_out
__global__ void gru_gate_kernel(const float* __restrict__ gi,
                                const float* __restrict__ gh,
                                const float* __restrict__ h0,
                                float* __restrict__ hnew,
                                float* __restrict__ out_h) {
  const int j = blockIdx.x * blockDim.x + threadIdx.x;
  if (j >= H_SZ) return;
  const float r = 1.f / (1.f + __expf(-(gi[j] + gh[j])));
  const float z = 1.f / (1.f + __expf(-(gi[H_SZ + j] + gh[H_SZ + j])));
  const float n = tanhf(gi[2 * H_SZ + j] + r * gh[2 * H_SZ + j]);
  const float h = (1.f - z) * n + z * h0[j];
  hnew[j]  = h;
  out_h[j] = h;
}

// single-block log_softmax over V=50000 (logits live in L2 after projection)
__global__ void log_softmax_kernel(const float* __restrict__ logits,
                                   float* __restrict__ out) {
  __shared__ float sm[1024];
  const int t = threadIdx.x;
  float mx = -INFINITY;
  for (int v = t; v < V_SZ; v += 1024) mx = fmaxf(mx, logits[v]);
  sm[t] = mx; __syncthreads();
  for (int o = 512; o > 0; o >>= 1) {
    if (t < o) sm[t] = fmaxf(sm[t], sm[t + o]);
    __syncthreads();
  }
  mx = sm[0]; __syncthreads();
  float s = 0.f;
  for (int v = t; v < V_SZ; v += 1024) s += __expf(logits[v] - mx);
  sm[t] = s; __syncthreads();
  for (int o = 512; o > 0; o >>= 1) {
    if (t < o) sm[t] += sm[t + o];
    __syncthreads();
  }
  const float lse = mx + logf(sm[0]);
  for (int v = t; v < V_SZ; v += 1024) out[v] = logits[v] - lse;
}

static inline void launch_matvec(const float* W, const float* x, const float* b,
                                 float* y, int R, int K, int act,
                                 hipStream_t stream) {
  const int tiles  = R >> 4;                 // R is a multiple of 16 everywhere
  const int blocks = (tiles + 3) / 4;        // 4 waves (128 threads) per block
  wmma_matvec_kernel<<<blocks, 128, 0, stream>>>(W, x, b, y, R, K, act);
}

extern "C" void kernel_launch(void* const* d_in, const int* in_sizes, int n_in,
                              void* d_out, int out_size, void* d_ws, size_t ws_size,
                              hipStream_t stream) {
  (void)in_sizes; (void)n_in; (void)out_size; (void)ws_size;

  const int*   input  = (const int*)  d_in[0];
  const float* hidden = (const float*)d_in[1];   // [1,1,H] -> H floats
  const float* enc    = (const float*)d_in[2];   // [L,H]
  const float* emb    = (const float*)d_in[3];   // [V,E]
  const float* attn_W = (const float*)d_in[4];   // [L,E+H]
  const float* attn_b = (const float*)d_in[5];
  const float* comb_W = (const float*)d_in[6];   // [H,E+H]
  const float* comb_b = (const float*)d_in[7];
  const float* W_ih   = (const float*)d_in[8];   // [3H,H]
  const float* W_hh   = (const float*)d_in[9];   // [3H,H]
  const float* b_ih   = (const float*)d_in[10];
  const float* b_hh   = (const float*)d_in[11];
  const float* out_W  = (const float*)d_in[12];  // [V,H]
  const float* out_b  = (const float*)d_in[13];

  float* ws        = (float*)d_ws;
  float* out       = (float*)d_out;
  float* out_logsm = out;                 // [V]
  float* out_h     = out + V_SZ;          // [H]
  float* out_attn  = out + V_SZ + H_SZ;   // [L]

  float* cat1   = ws + WS_CAT1;
  float* scores = ws + WS_SCORES;
  float* cat2   = ws + WS_CAT2;
  float* xvec   = ws + WS_X;
  float* gi     = ws + WS_GI;
  float* gh     = ws + WS_GH;
  float* hnew   = ws + WS_HNEW;
  float* logits = ws + WS_LOGITS;

  // 1. embedding gather + concat with h0
  build_cat1_kernel<<<10, 256, 0, stream>>>(input, emb, hidden, cat1, cat2);
  // 2. attention logits: [L] = attn_W [L, E+H] @ cat1
  launch_matvec(attn_W, cat1, attn_b, scores, L_SZ, EH_SZ, 0, stream);
  // 3. softmax over L
  softmax512_kernel<<<1, L_SZ, 0, stream>>>(scores, out_attn);
  // 4. context vector: cat2[E:] = w @ enc
  attn_apply_kernel<<<H_SZ / 256, 256, 0, stream>>>(scores, enc, cat2);
  // 5. combine + relu: x = relu(comb_W @ cat2 + comb_b)
  launch_matvec(comb_W, cat2, comb_b, xvec, H_SZ, EH_SZ, 1, stream);
  // 6. GRU matvecs + gates
  launch_matvec(W_ih, xvec,   b_ih, gi, 3 * H_SZ, H_SZ, 0, stream);
  launch_matvec(W_hh, hidden, b_hh, gh, 3 * H_SZ, H_SZ, 0, stream);
  gru_gate_kernel<<<H_SZ / 256, 256, 0, stream>>>(gi, gh, hidden, hnew, out_h);
  // 7. output projection: the 410 MB pass, 3125 WMMA tiles
  launch_matvec(out_W, hnew, out_b, logits, V_SZ, H_SZ, 0, stream);
  // 8. log_softmax over V
  log_softmax_kernel<<<1, 1024, 0, stream>>>(logits, out_logsm);
}